// GCN_33801392620006
// MI455X (gfx1250) — compile-verified
//
#include <hip/hip_runtime.h>
#include <hip/hip_bf16.h>

typedef __attribute__((ext_vector_type(2))) float v2f;
typedef __attribute__((ext_vector_type(8))) float v8f;

#define D 128   // D_IN == D_OUT == 128 for this problem

// ---------------------------------------------------------------------------
// deg[i] = 1 (self loop)
__global__ void k_init_deg(float* __restrict__ deg, int n) {
    int i = blockIdx.x * blockDim.x + threadIdx.x;
    if (i < n) deg[i] = 1.0f;
}

// deg[dst[e]] += 1
__global__ void k_deg(const int* __restrict__ dst, float* __restrict__ deg, int e) {
    int i = blockIdx.x * blockDim.x + threadIdx.x;
    if (i < e) atomicAdd(&deg[dst[i]], 1.0f);
}

// deg -> dinv = rsqrt(deg)   (deg >= 1 always, no zero guard needed)
__global__ void k_dinv(float* __restrict__ deg, int n) {
    int i = blockIdx.x * blockDim.x + threadIdx.x;
    if (i < n) deg[i] = rsqrtf(deg[i]);
}

// ---------------------------------------------------------------------------
// x_lin = x @ W^T  via V_WMMA_F32_16X16X4_F32.
// One wave -> one 16x16 output tile; K loop of 32 WMMAs.
// A[m][k]: lane (m = L&15, h = L>>4) holds {x[row][k+2h], x[row][k+2h+1]}.
// B[k][n] = W[n][k]: lane n holds {W[n][k+2h], W[n][k+2h+1]}.
// C/D: VGPR r, lane L -> row r+8h, col L&15.
__global__ void k_gemm_wmma(const float* __restrict__ x, const float* __restrict__ W,
                            float* __restrict__ xlin, int n, int nTiles) {
    int wave = blockIdx.x * (blockDim.x >> 5) + (threadIdx.x >> 5);
    if (wave >= nTiles) return;              // uniform per wave
    int rowTile = wave >> 3;                 // 8 col tiles (128/16)
    int colTile = wave & 7;
    int lane = threadIdx.x & 31;
    int m = lane & 15;
    int h = lane >> 4;

    int arowIdx = rowTile * 16 + m;
    if (arowIdx >= n) arowIdx = n - 1;       // clamp, keep EXEC full
    const float* arow = x + (size_t)arowIdx * D;
    const float* brow = W + (size_t)(colTile * 16 + m) * D;

    v8f c = {};
#pragma unroll
    for (int kk = 0; kk < D; kk += 4) {
        v2f a = *(const v2f*)(arow + kk + 2 * h);
        v2f b = *(const v2f*)(brow + kk + 2 * h);
        c = __builtin_amdgcn_wmma_f32_16x16x4_f32(
                /*neg_a=*/false, a, /*neg_b=*/false, b,
                /*c_mod=*/(short)0, c, /*reuse_a=*/false, /*reuse_b=*/false);
    }

    int row0 = rowTile * 16 + 8 * h;
    int col  = colTile * 16 + m;
#pragma unroll
    for (int r = 0; r < 8; ++r) {
        int row = row0 + r;
        if (row < n) xlin[(size_t)row * D + col] = c[r];
    }
}

// ---------------------------------------------------------------------------
// out[i] = b + dinv[i]^2 * x_lin[i]   (self-loop term + bias), float4 wide
__global__ void k_out_init(const float* __restrict__ xlin, const float* __restrict__ dinv,
                           const float* __restrict__ b, float* __restrict__ out, int n) {
    int idx = blockIdx.x * blockDim.x + threadIdx.x;   // n * 32 threads (D/4 per row)
    if (idx >= n * (D / 4)) return;
    int i = idx >> 5;
    int q = idx & 31;
    float s = dinv[i];
    s = s * s;
    float4 v  = ((const float4*)(xlin + (size_t)i * D))[q];
    float4 bv = ((const float4*)b)[q];
    float4 o;
    o.x = bv.x + v.x * s;
    o.y = bv.y + v.y * s;
    o.z = bv.z + v.z * s;
    o.w = bv.w + v.w * s;
    ((float4*)(out + (size_t)i * D))[q] = o;
}

// ---------------------------------------------------------------------------
// One wave per edge: out[dst] += dinv[src]*dinv[dst] * x_lin[src]
// Lane L handles float4 chunk L of the 128-wide row (coalesced gather; 4 atomics).
// x_lin (25.6 MB) is L2-resident on MI455X (192 MB L2) -> gathers/atomics stay on-chip.
__global__ void k_scatter(const float* __restrict__ xlin, const int* __restrict__ src,
                          const int* __restrict__ dst, const float* __restrict__ dinv,
                          float* __restrict__ out, int e) {
    int w = blockIdx.x * (blockDim.x >> 5) + (threadIdx.x >> 5);
    if (w >= e) return;                      // uniform per wave
    int lane = threadIdx.x & 31;
    int s = src[w];
    int d = dst[w];
    float nw = dinv[s] * dinv[d];
    float4 v = ((const float4*)(xlin + (size_t)s * D))[lane];
    float* od = out + (size_t)d * D + lane * 4;
    atomicAdd(od + 0, v.x * nw);
    atomicAdd(od + 1, v.y * nw);
    atomicAdd(od + 2, v.z * nw);
    atomicAdd(od + 3, v.w * nw);
}

// ---------------------------------------------------------------------------
extern "C" void kernel_launch(void* const* d_in, const int* in_sizes, int n_in,
                              void* d_out, int out_size, void* d_ws, size_t ws_size,
                              hipStream_t stream) {
    const float* x    = (const float*)d_in[0];   // [N, 128]
    const int*   eidx = (const int*)d_in[1];     // [2, E]
    const float* W    = (const float*)d_in[2];   // [128, 128]
    const float* b    = (const float*)d_in[3];   // [128]
    float*       out  = (float*)d_out;           // [N, 128]

    int N = in_sizes[0] / D;
    int E = in_sizes[1] / 2;
    const int* src = eidx;
    const int* dst = eidx + E;

    // workspace: [deg/dinv : N floats][x_lin : N*D floats]
    char* ws = (char*)d_ws;
    float* deg = (float*)ws;
    size_t off = (((size_t)N * sizeof(float)) + 255) & ~(size_t)255;
    float* xlin = (float*)(ws + off);

    k_init_deg<<<(N + 255) / 256, 256, 0, stream>>>(deg, N);
    k_deg<<<(E + 255) / 256, 256, 0, stream>>>(dst, deg, E);
    k_dinv<<<(N + 255) / 256, 256, 0, stream>>>(deg, N);

    int rowTiles = (N + 15) / 16;
    int nTiles = rowTiles * 8;                       // 8 col tiles of 16 over D_OUT=128
    k_gemm_wmma<<<(nTiles + 3) / 4, 128, 0, stream>>>(x, W, xlin, N, nTiles);

    int tInit = N * (D / 4);
    k_out_init<<<(tInit + 255) / 256, 256, 0, stream>>>(xlin, deg, b, out, N);

    k_scatter<<<(E + 7) / 8, 256, 0, stream>>>(xlin, src, dst, deg, out, E);
}